// RecurrentGCN_48644799594832
// MI455X (gfx1250) — compile-verified
//
#include <hip/hip_runtime.h>

typedef float v2f __attribute__((ext_vector_type(2)));
typedef float v8f __attribute__((ext_vector_type(8)));

#define IN_DIM 16
#define HID    32

// LDS k-pair-interleaved layouts: element (k,c) -> (k>>1)*PLD + 2*c + (k&1)
// PLD chosen even (b64 alignment) and ≡ 32 (mod 64) dwords so the upper lane
// half (k+2 -> next pair row) hits the opposite bank half: conflict-free b64.
#define WBIG_PLD 224   // 2*96 + 32 pad
#define WHH_PLD   96   // 2*32 + 32 pad
#define WLIN_PLD  32   // 2*16 (already ≡ 32 mod 64)
#define STAGE_LD  34   // even -> aligned b64 pair reads; conflict-free

// d_ws layout (float offsets)
#define WS_WBIG  0
#define WS_WHH   (48*96)
#define WS_WLIN  (48*96 + 32*32)
#define WS_TOTAL (48*96 + 32*32 + 32*16)

// ---------------------------------------------------------------------------
// Prep: fold w[0]+w[1] per gate into combined matrices (plain row-major in ws).
//   Wbig[48][96]: cols 0-31 = Wz, 32-63 = Wr, 64-95 = Wh (x-rows only, h-rows 0)
//   Whh [32][32]: Wh rows 16..47 (h-part of candidate gate)
//   Wlin[32][16]: w_lin zero-padded from 8 to 16 cols
// ---------------------------------------------------------------------------
__global__ void prep_weights(const float* __restrict__ wz,
                             const float* __restrict__ wr,
                             const float* __restrict__ wh,
                             const float* __restrict__ wlin,
                             float* __restrict__ ws) {
  int i = blockIdx.x * blockDim.x + threadIdx.x;
  const int S = 48 * 32;
  if (i < 48 * 96) {
    int k = i / 96, c = i % 96;
    float v;
    if (c < 32)      v = wz[k*32 + c]      + wz[S + k*32 + c];
    else if (c < 64) v = wr[k*32 + (c-32)] + wr[S + k*32 + (c-32)];
    else             v = (k < 16) ? (wh[k*32 + (c-64)] + wh[S + k*32 + (c-64)]) : 0.0f;
    ws[WS_WBIG + i] = v;
  } else if (i < 48*96 + 32*32) {
    int j = i - 48*96; int k = j / 32, c = j % 32;
    ws[i] = wh[(16 + k)*32 + c] + wh[S + (16 + k)*32 + c];
  } else if (i < WS_TOTAL) {
    int j = i - (48*96 + 32*32); int k = j / 16, c = j % 16;
    ws[i] = (c < 8) ? wlin[k*8 + c] : 0.0f;
  }
}

__device__ __forceinline__ float fast_sigmoid(float v) {
  return __builtin_amdgcn_rcpf(1.0f + __expf(-v));
}
__device__ __forceinline__ float fast_tanh(float v) {
  return 1.0f - 2.0f * __builtin_amdgcn_rcpf(__expf(2.0f * v) + 1.0f);
}

// ---------------------------------------------------------------------------
// Fused DCRNN-GRU: one wave32 per 16-node tile, f32 WMMA for all GEMMs.
// ---------------------------------------------------------------------------
__global__ __launch_bounds__(256) void dcrnn_kernel(
    const float* __restrict__ x, const float* __restrict__ h,
    const float* __restrict__ ws,
    const float* __restrict__ bz, const float* __restrict__ br,
    const float* __restrict__ bh, const float* __restrict__ blin,
    float* __restrict__ out, float* __restrict__ hnew_out,
    int N, int numTiles) {

  __shared__ float sWbig[24 * WBIG_PLD];   // 21.5 KB
  __shared__ float sWhh [16 * WHH_PLD];    //  6   KB
  __shared__ float sWlin[16 * WLIN_PLD];   //  2   KB
  __shared__ float sBias[3 * 32 + 8];
  __shared__ float sStage[8][16 * STAGE_LD];  // 17 KB, per-wave re-layout

  const int tid = threadIdx.x;
  // cooperative fill with k-pair interleaving
  for (int i = tid; i < 48*96; i += blockDim.x) {
    int k = i / 96, c = i % 96;
    sWbig[(k >> 1)*WBIG_PLD + 2*c + (k & 1)] = ws[WS_WBIG + i];
  }
  for (int i = tid; i < 32*32; i += blockDim.x) {
    int k = i / 32, c = i % 32;
    sWhh[(k >> 1)*WHH_PLD + 2*c + (k & 1)] = ws[WS_WHH + i];
  }
  for (int i = tid; i < 32*16; i += blockDim.x) {
    int k = i / 16, c = i % 16;
    sWlin[(k >> 1)*WLIN_PLD + 2*c + (k & 1)] = ws[WS_WLIN + i];
  }
  if (tid < 32)       sBias[tid] = bz[tid];
  else if (tid < 64)  sBias[tid] = br[tid - 32];
  else if (tid < 96)  sBias[tid] = bh[tid - 64];
  else if (tid < 104) sBias[tid] = blin[tid - 96];
  __syncthreads();

  const int wave = tid >> 5;
  const int lane = tid & 31;
  const int tile = blockIdx.x * 8 + wave;
  if (tile >= numTiles) return;          // wave-uniform: EXEC stays all-ones

  const int r0   = tile * 16;
  const int lrow = lane & 15;            // A-layout M / B,C-layout N
  const int half = lane >> 4;            // lane half -> K/M offset
  const int arow = min(r0 + lrow, N - 1);
  float* stg = sStage[wave];

  // ---- A tile: xh[16][48] in 16x4-f32 A-layout (lane holds K = kb, kb+1) ----
  v2f a[12];
  {
    const float* xr = x + (size_t)arow * IN_DIM;
    const float* hr = h + (size_t)arow * HID;
#pragma unroll
    for (int c = 0; c < 4; ++c)  a[c] = *(const v2f*)(xr + 4*c + 2*half);
#pragma unroll
    for (int c = 4; c < 12; ++c) a[c] = *(const v2f*)(hr + 4*c + 2*half - 16);
  }

  // ---- GEMM 1: [16,48] @ [48,96] -> z | r | candidate-x-part ----
  v8f acc[6];
#pragma unroll
  for (int t = 0; t < 6; ++t) { v8f zero = {}; acc[t] = zero; }
#pragma unroll
  for (int c = 0; c < 12; ++c) {
    const int p = 2*c + half;            // k-pair row for kb = 4c + 2*half
#pragma unroll
    for (int t = 0; t < 6; ++t) {
      const int col = t*16 + lrow;
      v2f b = *(const v2f*)&sWbig[p*WBIG_PLD + 2*col];  // one ds_load_b64
      acc[t] = __builtin_amdgcn_wmma_f32_16x16x4_f32(
          false, a[c], false, b, (short)0, acc[t], false, false);
    }
  }

  // ---- gates: sigmoid z, r; stage h*r in A-layout source buffer ----
  float zv[2][8], hv[2][8];
#pragma unroll
  for (int t = 0; t < 2; ++t) {
#pragma unroll
    for (int j = 0; j < 8; ++j) {
      const int m   = j + 8*half;               // C-layout row within tile
      const int col = t*16 + lrow;              // hidden index 0..31
      const int grow = min(r0 + m, N - 1);
      float zz = fast_sigmoid(acc[t    ][j] + sBias[col]);
      float rr = fast_sigmoid(acc[2 + t][j] + sBias[32 + col]);
      float hh = h[(size_t)grow * HID + col];
      zv[t][j] = zz;
      hv[t][j] = hh;
      stg[m*STAGE_LD + col] = hh * rr;          // same-wave LDS: in-order
    }
  }

  // ---- GEMM 2: (h*r)[16,32] @ Whh[32,32], C preloaded with x-part ----
  v8f acc2[2] = { acc[4], acc[5] };
#pragma unroll
  for (int c = 0; c < 8; ++c) {
    const int kb = 4*c + 2*half;
    v2f a2 = *(const v2f*)&stg[lrow*STAGE_LD + kb];     // aligned b64
    const int p = 2*c + half;
#pragma unroll
    for (int t = 0; t < 2; ++t) {
      const int col = t*16 + lrow;
      v2f b = *(const v2f*)&sWhh[p*WHH_PLD + 2*col];
      acc2[t] = __builtin_amdgcn_wmma_f32_16x16x4_f32(
          false, a2, false, b, (short)0, acc2[t], false, false);
    }
  }

  // ---- GRU blend, write h_new, stage relu(h_new) ----
#pragma unroll
  for (int t = 0; t < 2; ++t) {
#pragma unroll
    for (int j = 0; j < 8; ++j) {
      const int m   = j + 8*half;
      const int col = t*16 + lrow;
      const int grow = r0 + m;
      float htil = fast_tanh(acc2[t][j] + sBias[64 + col]);
      float z = zv[t][j];
      float hn = z * hv[t][j] + (1.0f - z) * htil;
      if (grow < N) hnew_out[(size_t)grow * HID + col] = hn;
      stg[m*STAGE_LD + col] = fmaxf(hn, 0.0f);
    }
  }

  // ---- GEMM 3: relu(h_new)[16,32] @ Wlin[32,16-padded] ----
  v8f acc3 = {};
#pragma unroll
  for (int c = 0; c < 8; ++c) {
    const int kb = 4*c + 2*half;
    v2f a3 = *(const v2f*)&stg[lrow*STAGE_LD + kb];
    const int p = 2*c + half;
    v2f b = *(const v2f*)&sWlin[p*WLIN_PLD + 2*lrow];
    acc3 = __builtin_amdgcn_wmma_f32_16x16x4_f32(
        false, a3, false, b, (short)0, acc3, false, false);
  }
  if (lrow < 8) {
#pragma unroll
    for (int j = 0; j < 8; ++j) {
      const int m = j + 8*half;
      const int grow = r0 + m;
      if (grow < N) out[(size_t)grow * 8 + lrow] = acc3[j] + sBias[96 + lrow];
    }
  }
}

// ---------------------------------------------------------------------------
extern "C" void kernel_launch(void* const* d_in, const int* in_sizes, int n_in,
                              void* d_out, int out_size, void* d_ws, size_t ws_size,
                              hipStream_t stream) {
  (void)n_in; (void)out_size; (void)ws_size;
  const float* x    = (const float*)d_in[0];
  // d_in[1] = edge_index, d_in[2] = edge_weight: dead for K=1 (skip 96 MB of reads)
  const float* h    = (const float*)d_in[3];
  const float* wz   = (const float*)d_in[4];
  const float* bz   = (const float*)d_in[5];
  const float* wr   = (const float*)d_in[6];
  const float* br   = (const float*)d_in[7];
  const float* wh   = (const float*)d_in[8];
  const float* bh   = (const float*)d_in[9];
  const float* wlin = (const float*)d_in[10];
  const float* blin = (const float*)d_in[11];

  float* out = (float*)d_out;
  float* ws  = (float*)d_ws;
  const int N = in_sizes[0] / IN_DIM;

  prep_weights<<<(WS_TOTAL + 255) / 256, 256, 0, stream>>>(wz, wr, wh, wlin, ws);

  const int numTiles = (N + 15) / 16;
  const int blocks   = (numTiles + 7) / 8;   // 8 waves (tiles) per 256-thread block
  dcrnn_kernel<<<blocks, 256, 0, stream>>>(
      x, h, ws, bz, br, bh, blin,
      out, out + (size_t)N * 8, N, numTiles);
}